// SentinelFlowGCN_4758823764428
// MI455X (gfx1250) — compile-verified
//
#include <hip/hip_runtime.h>
#include <hip/hip_bf16.h>
#include <math.h>

typedef __attribute__((ext_vector_type(2))) float v2f;
typedef __attribute__((ext_vector_type(8))) float v8f;

#define EPS_BN 1e-5f

// ---------------------------------------------------------------------------
// Degree: deg[dst] += 1 per edge (float atomics; deg later becomes dinv)
// ---------------------------------------------------------------------------
__global__ __launch_bounds__(256) void deg_count_kernel(
    const long long* __restrict__ edst, float* __restrict__ deg, long long ne) {
    long long e = (long long)blockIdx.x * blockDim.x + threadIdx.x;
    if (e < ne) atomicAdd(&deg[(int)edst[e]], 1.0f);
}

__global__ __launch_bounds__(256) void deg_to_dinv_kernel(
    float* __restrict__ deg, int n) {
    int i = blockIdx.x * blockDim.x + threadIdx.x;
    if (i < n) deg[i] = rsqrtf(deg[i] + 1.0f);   // +1 self loop; deg >= 1
}

// ---------------------------------------------------------------------------
// Dense transform H[N,ODIM] = X[N,KDIM] @ W[KDIM,ODIM] using
// V_WMMA_F32_16X16X4_F32. One wave computes a 16 x ODIM row-block.
// Requires nrows % 16 == 0 (N = 100000 = 6250*16) and full EXEC per wave.
// ---------------------------------------------------------------------------
template <int KDIM, int ODIM>
__global__ __launch_bounds__(256) void gcn_linear_wmma_kernel(
    const float* __restrict__ X, const float* __restrict__ W,
    float* __restrict__ H, int nrows) {
    const int lane  = threadIdx.x & 31;
    const int wave  = blockIdx.x * (blockDim.x >> 5) + (threadIdx.x >> 5);
    const int mtile = nrows >> 4;
    if (wave >= mtile) return;               // uniform per-wave exit (EXEC stays all-1)

    constexpr int NT = ODIM / 16;
    v8f acc[NT] = {};

    const int m  = lane & 15;                // row / col within 16-tile
    const int kk = (lane >> 4) << 1;         // K sub-offset {0,2}
    const size_t row0 = (size_t)wave << 4;

    for (int k0 = 0; k0 < KDIM; k0 += 4) {
        // A fragment: 16x4 fp32. VGPR0: K=k0+kk, VGPR1: K=k0+kk+1, row = lane&15
        v2f a;
        a.x = X[(row0 + m) * KDIM + (k0 + kk)];
        a.y = X[(row0 + m) * KDIM + (k0 + kk + 1)];
#pragma unroll
        for (int t = 0; t < NT; ++t) {
            // B fragment: 4x16 fp32. Row K striped across lanes (coalesced on W).
            v2f b;
            b.x = W[(size_t)(k0 + kk) * ODIM + t * 16 + m];
            b.y = W[(size_t)(k0 + kk + 1) * ODIM + t * 16 + m];
            acc[t] = __builtin_amdgcn_wmma_f32_16x16x4_f32(
                /*neg_a=*/false, a, /*neg_b=*/false, b,
                /*c_mod=*/(short)0, acc[t], /*reuse_a=*/false, /*reuse_b=*/false);
        }
    }

    // D layout: VGPR r, lane l -> row = r + 8*(l>>4), col = l&15
    const int rbase = (lane >> 4) * 8;
#pragma unroll
    for (int t = 0; t < NT; ++t) {
#pragma unroll
        for (int r = 0; r < 8; ++r) {
            H[(row0 + rbase + r) * ODIM + t * 16 + m] = acc[t][r];
        }
    }
}

// ---------------------------------------------------------------------------
// Edge scatter: agg[dst] += h[src] * dinv[src]*dinv[dst]
// One thread handles one float4 chunk of one edge's features.
// ---------------------------------------------------------------------------
template <int ODIM>
__global__ __launch_bounds__(256) void gcn_scatter_kernel(
    const float* __restrict__ H, const long long* __restrict__ esrc,
    const long long* __restrict__ edst, const float* __restrict__ dinv,
    float* __restrict__ agg, long long ne) {
    constexpr int QPE = ODIM / 4;            // float4 chunks per edge
    long long gid = (long long)blockIdx.x * blockDim.x + threadIdx.x;
    long long e = gid / QPE;
    if (e >= ne) return;
    int q = (int)(gid - e * QPE);
    int s = (int)esrc[e];
    int d = (int)edst[e];
    float nrm = dinv[s] * dinv[d];
    const float4 v = *(const float4*)(H + (size_t)s * ODIM + q * 4);
    float* ap = agg + (size_t)d * ODIM + q * 4;
    atomicAdd(ap + 0, v.x * nrm);
    atomicAdd(ap + 1, v.y * nrm);
    atomicAdd(ap + 2, v.z * nrm);
    atomicAdd(ap + 3, v.w * nrm);
}

// ---------------------------------------------------------------------------
// Finalize: out = relu(BN(agg + h*dinv^2 + b))
// ---------------------------------------------------------------------------
template <int ODIM>
__global__ __launch_bounds__(256) void gcn_finalize_kernel(
    const float* __restrict__ agg, const float* __restrict__ H,
    const float* __restrict__ dinv, const float* __restrict__ bias,
    const float* __restrict__ g, const float* __restrict__ be,
    const float* __restrict__ rm, const float* __restrict__ rv,
    float* __restrict__ out, int n) {
    long long idx = (long long)blockIdx.x * blockDim.x + threadIdx.x;
    if (idx >= (long long)n * ODIM) return;
    int node = (int)(idx / ODIM);
    int f    = (int)(idx - (long long)node * ODIM);
    float di = dinv[node];
    float val = agg[idx] + H[idx] * di * di + bias[f];
    val = (val - rm[f]) * rsqrtf(rv[f] + EPS_BN) * g[f] + be[f];
    out[idx] = fmaxf(val, 0.0f);
}

// ---------------------------------------------------------------------------
// Classifier: logits = h3 @ Wc + bc ; log_softmax over 2 classes
// ---------------------------------------------------------------------------
__global__ __launch_bounds__(256) void classifier_kernel(
    const float* __restrict__ X, const float* __restrict__ Wc,
    const float* __restrict__ bc, float* __restrict__ out, int n) {
    int node = blockIdx.x * blockDim.x + threadIdx.x;
    if (node >= n) return;
    float l0 = bc[0], l1 = bc[1];
    const float* xp = X + (size_t)node * 32;
#pragma unroll
    for (int k = 0; k < 32; ++k) {
        float x = xp[k];
        l0 = fmaf(x, Wc[k * 2 + 0], l0);
        l1 = fmaf(x, Wc[k * 2 + 1], l1);
    }
    float mx  = fmaxf(l0, l1);
    float lse = mx + logf(expf(l0 - mx) + expf(l1 - mx));
    out[(size_t)node * 2 + 0] = l0 - lse;
    out[(size_t)node * 2 + 1] = l1 - lse;
}

// ---------------------------------------------------------------------------
static inline size_t align256(size_t x) { return (x + 255) & ~(size_t)255; }
static inline unsigned cdivu(long long a, long long b) { return (unsigned)((a + b - 1) / b); }

extern "C" void kernel_launch(void* const* d_in, const int* in_sizes, int n_in,
                              void* d_out, int out_size, void* d_ws, size_t ws_size,
                              hipStream_t stream) {
    const int       N = in_sizes[0] / 8;            // x is [N,8]
    const long long E = (long long)(in_sizes[1] / 2); // edge_index is [2,E] int64

    const float*     x    = (const float*)d_in[0];
    const long long* eidx = (const long long*)d_in[1];
    const long long* esrc = eidx;        // edge_index[0]
    const long long* edst = eidx + E;    // edge_index[1]

    const float* W1 = (const float*)d_in[2];  const float* b1 = (const float*)d_in[3];
    const float* g1 = (const float*)d_in[4];  const float* be1= (const float*)d_in[5];
    const float* rm1= (const float*)d_in[6];  const float* rv1= (const float*)d_in[7];
    const float* W2 = (const float*)d_in[8];  const float* b2 = (const float*)d_in[9];
    const float* g2 = (const float*)d_in[10]; const float* be2= (const float*)d_in[11];
    const float* rm2= (const float*)d_in[12]; const float* rv2= (const float*)d_in[13];
    const float* W3 = (const float*)d_in[14]; const float* b3 = (const float*)d_in[15];
    const float* g3 = (const float*)d_in[16]; const float* be3= (const float*)d_in[17];
    const float* rm3= (const float*)d_in[18]; const float* rv3= (const float*)d_in[19];
    const float* Wc = (const float*)d_in[20]; const float* bc = (const float*)d_in[21];

    // Workspace carve-up (all 256B aligned)
    char*  ws   = (char*)d_ws;
    float* dinv = (float*)ws;                 ws += align256((size_t)N * 4);
    float* h    = (float*)ws;                 ws += align256((size_t)N * 64 * 4);
    float* agg  = (float*)ws;                 ws += align256((size_t)N * 64 * 4);
    float* xcur = (float*)ws;                 ws += align256((size_t)N * 64 * 4);

    const int TB = 256;

    // --- degree -> dinv -------------------------------------------------
    hipMemsetAsync(dinv, 0, (size_t)N * 4, stream);
    deg_count_kernel<<<cdivu(E, TB), TB, 0, stream>>>(edst, dinv, E);
    deg_to_dinv_kernel<<<cdivu(N, TB), TB, 0, stream>>>(dinv, N);

    const unsigned gemm_blocks = cdivu((long long)(N / 16), TB / 32);

    // --- Layer 1: 8 -> 64 ----------------------------------------------
    gcn_linear_wmma_kernel<8, 64><<<gemm_blocks, TB, 0, stream>>>(x, W1, h, N);
    hipMemsetAsync(agg, 0, (size_t)N * 64 * 4, stream);
    gcn_scatter_kernel<64><<<cdivu(E * 16, TB), TB, 0, stream>>>(h, esrc, edst, dinv, agg, E);
    gcn_finalize_kernel<64><<<cdivu((long long)N * 64, TB), TB, 0, stream>>>(
        agg, h, dinv, b1, g1, be1, rm1, rv1, xcur, N);

    // --- Layer 2: 64 -> 64 ----------------------------------------------
    gcn_linear_wmma_kernel<64, 64><<<gemm_blocks, TB, 0, stream>>>(xcur, W2, h, N);
    hipMemsetAsync(agg, 0, (size_t)N * 64 * 4, stream);
    gcn_scatter_kernel<64><<<cdivu(E * 16, TB), TB, 0, stream>>>(h, esrc, edst, dinv, agg, E);
    gcn_finalize_kernel<64><<<cdivu((long long)N * 64, TB), TB, 0, stream>>>(
        agg, h, dinv, b2, g2, be2, rm2, rv2, xcur, N);

    // --- Layer 3: 64 -> 32 ----------------------------------------------
    gcn_linear_wmma_kernel<64, 32><<<gemm_blocks, TB, 0, stream>>>(xcur, W3, h, N);
    hipMemsetAsync(agg, 0, (size_t)N * 32 * 4, stream);
    gcn_scatter_kernel<32><<<cdivu(E * 8, TB), TB, 0, stream>>>(h, esrc, edst, dinv, agg, E);
    gcn_finalize_kernel<32><<<cdivu((long long)N * 32, TB), TB, 0, stream>>>(
        agg, h, dinv, b3, g3, be3, rm3, rv3, xcur, N);

    // --- Classifier + log_softmax ---------------------------------------
    classifier_kernel<<<cdivu(N, TB), TB, 0, stream>>>(xcur, Wc, bc, (float*)d_out, N);
}